// PointerDecoder_56032143344273
// MI455X (gfx1250) — compile-verified
//
#include <hip/hip_runtime.h>
#include <math.h>

#define B_  128
#define N_  1024
#define DIN 512
#define D_  512
#define DCTX 1024
#define H_  8
#define HD_ 64

typedef __attribute__((ext_vector_type(2))) float v2f;
typedef __attribute__((ext_vector_type(8))) float v8f;

__device__ __forceinline__ v8f wmma4(v2f a, v2f b, v8f c) {
  // V_WMMA_F32_16X16X4_F32: A 16x4 f32 (2 VGPR), B 4x16 f32 (2 VGPR), C/D 16x16 f32 (8 VGPR)
  return __builtin_amdgcn_wmma_f32_16x16x4_f32(false, a, false, b, (short)0, c, false, false);
}

// ---------------------------------------------------------------------------
// Pass 0: S[b,d] = sum_n emb[b,n,d]   (pure bandwidth, 268 MB read)
// ---------------------------------------------------------------------------
__global__ void sum_kernel(const float* __restrict__ emb, float* __restrict__ S) {
  int b = blockIdx.y;
  int d = blockIdx.x * 128 + threadIdx.x;
  const float* p = emb + (size_t)b * N_ * DIN + d;
  float a0 = 0.f, a1 = 0.f, a2 = 0.f, a3 = 0.f;
  for (int n = 0; n < N_; n += 4) {
    a0 += p[(size_t)(n + 0) * DIN];
    a1 += p[(size_t)(n + 1) * DIN];
    a2 += p[(size_t)(n + 2) * DIN];
    a3 += p[(size_t)(n + 3) * DIN];
  }
  S[b * DIN + d] = (a0 + a1) + (a2 + a3);
}

// ---------------------------------------------------------------------------
// Generic f32 WMMA GEMM, compile-time specialized (branch-free inner loop):
//   C[m,n] = scale * (sum_k A[m,k]*B[k,n] + (ACC ? Cin : 0) + (BIAS ? bias[n] : 0))
// One wave per 16x16 tile; K unrolled by 4 -> 4 consecutive v_wmma per trip.
// BTRANS: B[k,n] = Bp[n*ldb + k]  (else Bp[k*ldb + n])
// ---------------------------------------------------------------------------
template <bool BTRANS, bool ACC, bool BIAS>
__launch_bounds__(32)
__global__ void gemm_f32_wmma(const float* __restrict__ A, int lda,
                              const float* __restrict__ Bp, int ldb,
                              const float* __restrict__ Cin, float* __restrict__ Cout,
                              long ldc_row, long ldc_col, long c_off,
                              int Ntiles, int K,
                              const float* __restrict__ bias, float scale) {
  int tile = blockIdx.x;
  int m0 = (tile / Ntiles) * 16;
  int n0 = (tile % Ntiles) * 16;
  int lane = threadIdx.x;
  int l16 = lane & 15;
  int kh  = lane >> 4;  // 0: holds K+{0,1}, D rows 0-7 ; 1: holds K+{2,3}, D rows 8-15

  v8f c = {};
  if (ACC) {
    #pragma unroll
    for (int j = 0; j < 8; ++j)
      c[j] = Cin[c_off + (long)(m0 + j + 8 * kh) * ldc_row + (long)(n0 + l16) * ldc_col];
  }

  const float* ap = A + (size_t)(m0 + l16) * lda + 2 * kh;
  const float* bpT = Bp + (size_t)(n0 + l16) * ldb + 2 * kh;   // BTRANS path
  const float* bpN = Bp + (size_t)(2 * kh) * ldb + n0 + l16;   // normal path

  for (int k0 = 0; k0 < K; k0 += 16) {
    #pragma unroll
    for (int u = 0; u < 4; ++u) {
      int k = k0 + 4 * u;
      v2f a = *(const v2f*)(ap + k);
      v2f bb;
      if (BTRANS) {
        bb.x = bpT[k];
        bb.y = bpT[k + 1];
      } else {
        bb.x = bpN[(size_t)k * ldb];
        bb.y = bpN[(size_t)k * ldb + ldb];
      }
      c = wmma4(a, bb, c);
    }
  }

  #pragma unroll
  for (int j = 0; j < 8; ++j) {
    int row = m0 + j + 8 * kh;
    int col = n0 + l16;
    float v = c[j];
    if (BIAS) v += bias[col];
    v *= scale;
    Cout[c_off + (long)row * ldc_row + (long)col * ldc_col] = v;
  }
}

// ---------------------------------------------------------------------------
// Fused attention: one block (8 waves) per batch.
// Phase 1: scores[n,h] = emb[b,n,:]·u[b,h,:] + s0[h]   (WMMA, heads padded to 16)
// Phase 2: per-head softmax over N (wave-level shuffles)
// Phase 3: w[h,:] = sum_n p[n,h]*emb[b,n,:]            (WMMA)
// Phase 4: ctx = w·wv + bv ; Q2 = ctx·wo + bo          (VALU, tiny)
// ---------------------------------------------------------------------------
__launch_bounds__(256)
__global__ void attn_kernel(const float* __restrict__ emb,
                            const unsigned char* __restrict__ mask,
                            const float* __restrict__ qh,
                            const float* __restrict__ u_pad,
                            const float* __restrict__ bk,
                            const float* __restrict__ wv,
                            const float* __restrict__ bv,
                            const float* __restrict__ wo,
                            const float* __restrict__ bo,
                            float* __restrict__ Q2out) {
  __shared__ float s_lds[N_ * 8];    // 32 KB: scores -> exp weights
  __shared__ float w_lds[8 * DIN];   // 16 KB
  __shared__ float ctx_lds[512];
  __shared__ float s0_lds[8];
  __shared__ float invl_lds[8];

  int b = blockIdx.x;
  int t = threadIdx.x;
  int wave = t >> 5, lane = t & 31, l16 = lane & 15, kh = lane >> 4;
  const float* embB = emb + (size_t)b * N_ * DIN;

  if (t < 8) {  // bias fold: s0[h] = q[b,h,:]·bk[h,:]
    float s = 0.f;
    for (int k = 0; k < HD_; ++k) s += qh[b * D_ + t * HD_ + k] * bk[t * HD_ + k];
    s0_lds[t] = s;
  }
  __syncthreads();

  // ---- Phase 1: scores via WMMA, A = emb tile (16 rows), B = u_pad (DINx16)
  for (int tile = wave; tile < N_ / 16; tile += 8) {
    int n0 = tile * 16;
    const float* ap = embB + (size_t)(n0 + l16) * DIN + 2 * kh;
    const float* up0 = u_pad + ((size_t)b * DIN + 2 * kh) * 16 + l16;
    // prefetch this wave's next tile (8 tiles / 128 rows ahead)
    if (tile + 8 < N_ / 16)
      __builtin_prefetch(embB + (size_t)(n0 + 128 + l16) * DIN, 0, 1);
    v8f c = {};
    for (int k0 = 0; k0 < DIN; k0 += 16) {
      #pragma unroll
      for (int u = 0; u < 4; ++u) {
        int k = k0 + 4 * u;
        v2f a = *(const v2f*)(ap + k);
        v2f bb;
        bb.x = up0[(size_t)k * 16];
        bb.y = up0[(size_t)k * 16 + 16];
        c = wmma4(a, bb, c);
      }
    }
    if (l16 < 8) {
      #pragma unroll
      for (int j = 0; j < 8; ++j) {
        int n = n0 + j + 8 * kh;
        float v = c[j] + s0_lds[l16];
        if (!mask[b * N_ + n]) v = -3.4028235e38f;   // flax: big_neg where mask==False
        s_lds[n * 8 + l16] = v;
      }
    }
  }
  __syncthreads();

  // ---- Phase 2: softmax per head; 32 threads (one wave) per head
  {
    int h = t >> 5, j = t & 31;
    float m = -3.4028235e38f;
    for (int n = j; n < N_; n += 32) m = fmaxf(m, s_lds[n * 8 + h]);
    #pragma unroll
    for (int o = 16; o > 0; o >>= 1) m = fmaxf(m, __shfl_xor(m, o, 32));
    float l = 0.f;
    for (int n = j; n < N_; n += 32) {
      float p = __expf(s_lds[n * 8 + h] - m);
      s_lds[n * 8 + h] = p;
      l += p;
    }
    #pragma unroll
    for (int o = 16; o > 0; o >>= 1) l += __shfl_xor(l, o, 32);
    if (j == 0) invl_lds[h] = 1.f / l;
  }
  __syncthreads();

  // ---- Phase 3: w[h,d] = sum_n p[n,h] * emb[b,n,d] via WMMA (A = p^T, M=16 pad)
  for (int dt = wave; dt < DIN / 16; dt += 8) {
    int d0 = dt * 16;
    const float* bp = embB + (size_t)(2 * kh) * DIN + d0 + l16;
    const float* pp = s_lds + (2 * kh) * 8 + ((l16 < 8) ? l16 : 0);
    float amask = (l16 < 8) ? 1.f : 0.f;
    v8f c = {};
    for (int k0 = 0; k0 < N_; k0 += 8) {
      #pragma unroll
      for (int u = 0; u < 2; ++u) {
        int n = k0 + 4 * u;
        v2f a;
        a.x = amask * pp[(size_t)n * 8];
        a.y = amask * pp[(size_t)n * 8 + 8];
        v2f bb;
        bb.x = bp[(size_t)n * DIN];
        bb.y = bp[(size_t)n * DIN + DIN];
        c = wmma4(a, bb, c);
      }
    }
    if (kh == 0) {  // lanes 0-15 hold D rows M=0..7 == real heads
      #pragma unroll
      for (int j = 0; j < 8; ++j)
        w_lds[j * DIN + d0 + l16] = c[j] * invl_lds[j];
    }
  }
  __syncthreads();

  // ---- Phase 4a: ctx[h,k] = w[h,:]·wv[:,h,k] + bv[h,k]
  for (int e = t; e < 512; e += 256) {
    int h = e >> 6, k = e & 63;
    float acc = bv[h * HD_ + k];
    const float* wl = w_lds + h * DIN;
    for (int i = 0; i < DIN; ++i) acc += wl[i] * wv[(size_t)i * 512 + h * HD_ + k];
    ctx_lds[e] = acc;
  }
  __syncthreads();
  // ---- Phase 4b: Q2[d] = sum_{h,k} ctx[h,k]*wo[h,k,d] + bo[d]
  for (int d = t; d < D_; d += 256) {
    float acc = bo[d];
    for (int j = 0; j < 512; ++j) acc += ctx_lds[j] * wo[(size_t)j * D_ + d];
    Q2out[(size_t)b * D_ + d] = acc;
  }
}

// ---------------------------------------------------------------------------
// Final pass: logits[b,n] = mask ? -inf : 10*tanh( (emb[b,n,:]·t[b,:]) / sqrt(512) )
// WMMA with rank-1 B (only column 0 live).
// ---------------------------------------------------------------------------
__launch_bounds__(256)
__global__ void logits_kernel(const float* __restrict__ emb,
                              const float* __restrict__ tvec,
                              const unsigned char* __restrict__ mask,
                              float* __restrict__ out) {
  int b = blockIdx.x;
  int t = threadIdx.x, wave = t >> 5, lane = t & 31, l16 = lane & 15, kh = lane >> 4;
  const float* embB = emb + (size_t)b * N_ * DIN;
  const float* tb = tvec + (size_t)b * DIN + 2 * kh;
  const float bmask = (l16 == 0) ? 1.f : 0.f;
  const float rs = 0.04419417382415922f;  // 1/sqrt(512)
  for (int tile = wave; tile < N_ / 16; tile += 8) {
    int n0 = tile * 16;
    const float* ap = embB + (size_t)(n0 + l16) * DIN + 2 * kh;
    if (tile + 8 < N_ / 16)
      __builtin_prefetch(embB + (size_t)(n0 + 128 + l16) * DIN, 0, 1);
    v8f c = {};
    for (int k0 = 0; k0 < DIN; k0 += 16) {
      #pragma unroll
      for (int u = 0; u < 4; ++u) {
        int k = k0 + 4 * u;
        v2f a = *(const v2f*)(ap + k);
        v2f bb;
        bb.x = bmask * tb[k];
        bb.y = bmask * tb[k + 1];
        c = wmma4(a, bb, c);
      }
    }
    if (l16 == 0) {  // column 0 of D
      #pragma unroll
      for (int j = 0; j < 8; ++j) {
        int n = n0 + j + 8 * kh;
        float v = 10.f * tanhf(c[j] * rs);
        if (mask[b * N_ + n]) v = -__builtin_inff();
        out[(size_t)b * N_ + n] = v;
      }
    }
  }
}

// ---------------------------------------------------------------------------
extern "C" void kernel_launch(void* const* d_in, const int* in_sizes, int n_in,
                              void* d_out, int out_size, void* d_ws, size_t ws_size,
                              hipStream_t stream) {
  const float* emb  = (const float*)d_in[0];
  const float* sctx = (const float*)d_in[1];
  const unsigned char* mask = (const unsigned char*)d_in[2];  // bool tensor
  const float* Wq_fixed = (const float*)d_in[3];
  const float* Wq_step  = (const float*)d_in[4];
  const float* Wk       = (const float*)d_in[5];
  const float* mha_wq = (const float*)d_in[6];
  const float* mha_bq = (const float*)d_in[7];
  const float* mha_wk = (const float*)d_in[8];
  const float* mha_bk = (const float*)d_in[9];
  const float* mha_wv = (const float*)d_in[10];
  const float* mha_bv = (const float*)d_in[11];
  const float* mha_wo = (const float*)d_in[12];
  const float* mha_bo = (const float*)d_in[13];
  float* out = (float*)d_out;

  float* ws = (float*)d_ws;
  float* S     = ws;                    // 128*512
  float* Q1    = S     + B_ * DIN;      // 128*512
  float* qh    = Q1    + B_ * D_;       // 128*512 (scaled q, heads flat)
  float* Q2    = qh    + B_ * D_;       // 128*512
  float* tv    = Q2    + B_ * D_;       // 128*512
  float* u_pad = tv    + B_ * DIN;      // 128*512*16 (heads padded to 16)

  // Pass 0: per-batch embedding sum
  sum_kernel<<<dim3(DIN / 128, B_), 128, 0, stream>>>(emb, S);

  // Q1 = step_context @ Wq_step      [128x1024]x[1024x512]
  gemm_f32_wmma<false, false, false><<<dim3((B_ / 16) * (D_ / 16)), 32, 0, stream>>>(
      sctx, DCTX, Wq_step, D_, nullptr, Q1, D_, 1, 0, D_ / 16, DCTX, nullptr, 1.f);
  // Q1 += S @ Wq_fixed               [128x512]x[512x512]
  gemm_f32_wmma<false, true, false><<<dim3((B_ / 16) * (D_ / 16)), 32, 0, stream>>>(
      S, DIN, Wq_fixed, D_, Q1, Q1, D_, 1, 0, D_ / 16, DIN, nullptr, 1.f);
  // qh = (Q1 @ mha_wq + bq) / sqrt(HD)
  gemm_f32_wmma<false, false, true><<<dim3((B_ / 16) * (D_ / 16)), 32, 0, stream>>>(
      Q1, D_, mha_wq, D_, nullptr, qh, D_, 1, 0, D_ / 16, D_, mha_bq, 0.125f);
  // u[b,:,h] = qh_h[b,:] @ wk_h^T : per-head [128x64]x[64x512], B transposed
  for (int h = 0; h < H_; ++h) {
    gemm_f32_wmma<true, false, false><<<dim3((B_ / 16) * (DIN / 16)), 32, 0, stream>>>(
        qh + h * HD_, D_, mha_wk + h * HD_, D_, nullptr, u_pad,
        (long)DIN * 16, 16, h, DIN / 16, HD_, nullptr, 1.f);
  }

  // Fused attention -> Q2
  attn_kernel<<<B_, 256, 0, stream>>>(emb, mask, qh, u_pad, mha_bk, mha_wv,
                                      mha_bv, mha_wo, mha_bo, Q2);

  // t[b,:] = Wk @ Q2[b,:]  -> GEMM [128x512]x[512x512] with B transposed
  gemm_f32_wmma<true, false, false><<<dim3((B_ / 16) * (DIN / 16)), 32, 0, stream>>>(
      Q2, D_, Wk, D_, nullptr, tv, DIN, 1, 0, DIN / 16, D_, nullptr, 1.f);

  // Final pass: logits
  logits_kernel<<<B_, 256, 0, stream>>>(emb, tv, mask, out);
}